// BatchedCauchyKernel3d_3891240370726
// MI455X (gfx1250) — compile-verified
//
#include <hip/hip_runtime.h>

// ---------------------------------------------------------------------------
// BatchedCauchyKernel3d for MI455X (gfx1250, wave32)
//
// out[b,m,n] = sxy / (sxy + d),
//   d   = clip(|x_m|^2 + |y_n|^2 - 2 x_m.y_n, 1e-10, 1e6)
//   sxy = sqrt(clip(scale_x[m]*scale_y[n], 1e-10, 1e12))
//       = clamp(rsx[m]*rsy[n], 1e-5, 1e6), rs* = sqrt(clip(scale_*,1e-10,1e6))
//
// Store-bandwidth bound: 256 MB out / 23.3 TB/s ~= 11 us floor.
// K=3 dot via V_WMMA_F32_16X16X4_F32 (K padded to 4 with zeros; the -2 of
// "-2 x.y" is folded into the A operand since f32 WMMA has no A/B-neg).
// Operand construction is branchless (per-lane addressing + v_cndmask) so
// EXEC stays all-ones through the loop (WMMA requirement, no exec save/restore).
// ---------------------------------------------------------------------------

typedef __attribute__((ext_vector_type(2))) float v2f;
typedef __attribute__((ext_vector_type(8))) float v8f;

#define B_   4
#define NX_  4096
#define NY_  4096
#define F_   16
#define BN_  (B_ * NX_)             // 16384 points per side (B*NX == B*NY)
#define NT_  8                      // 16-col n-tiles per wave strip
#define NSTRIPS_ (NY_ / (16 * NT_)) // 32 strips across N

// ws layout (interleaved pairs so consumers get vector loads):
//   wsx[2t]   = |x_t|^2        wsx[2t+1] = sqrt(clip(sample_x_t . s))
//   wsy[2t]   = |y_t|^2        wsy[2t+1] = sqrt(clip(sample_y_t . s))
// wsx = ws, wsy = ws + 2*BN_.  Total 4*BN_*4 = 256 KB.

__global__ __launch_bounds__(256) void cauchy_precompute(
    const float* __restrict__ x, const float* __restrict__ y,
    const float* __restrict__ sx, const float* __restrict__ sy,
    const float* __restrict__ scale, float* __restrict__ ws)
{
  int t = blockIdx.x * blockDim.x + threadIdx.x;
  if (t >= BN_) return;

  float s[F_];
#pragma unroll
  for (int f = 0; f < F_; ++f)
    s[f] = fminf(fmaxf(scale[f], 1e-6f), 1e6f);

  const float* xp = x + (size_t)t * 3;
  float sqx = xp[0] * xp[0] + xp[1] * xp[1] + xp[2] * xp[2];
  const float* sxp = sx + (size_t)t * F_;
  float ax = 0.0f;
#pragma unroll
  for (int f = 0; f < F_; ++f) ax += sxp[f] * s[f];
  ax = fminf(fmaxf(ax, 1e-10f), 1e6f);
  ws[2 * t]     = sqx;
  ws[2 * t + 1] = __builtin_amdgcn_sqrtf(ax);

  const float* yp = y + (size_t)t * 3;
  float sqy = yp[0] * yp[0] + yp[1] * yp[1] + yp[2] * yp[2];
  const float* syp = sy + (size_t)t * F_;
  float ay = 0.0f;
#pragma unroll
  for (int f = 0; f < F_; ++f) ay += syp[f] * s[f];
  ay = fminf(fmaxf(ay, 1e-10f), 1e6f);
  ws[2 * BN_ + 2 * t]     = sqy;
  ws[2 * BN_ + 2 * t + 1] = __builtin_amdgcn_sqrtf(ay);
}

// One wave per 16x(16*NT_) output strip. A operand (x-tile, scaled by -2) is
// built once per wave; per n-tile: one b64 ws load + 2 y loads + 1 WMMA +
// 8x(2 med3, rcp, fma) + 8 streaming b32 stores.
__global__ __launch_bounds__(256) void cauchy_wmma(
    const float* __restrict__ x, const float* __restrict__ y,
    const float* __restrict__ ws, float* __restrict__ out)
{
  const float* wsx = ws;
  const float* wsy = ws + 2 * BN_;

  int wave = (blockIdx.x * 256 + (int)threadIdx.x) >> 5;
  int lane = threadIdx.x & 31;
  int half = lane >> 4;      // 0 -> K=0/1, 1 -> K=2/3
  int l    = lane & 15;      // M (for A) / N (for B,C,D) within tile

  int ns = wave & (NSTRIPS_ - 1);
  int mt = (wave / NSTRIPS_) & (NX_ / 16 - 1);
  int b  = wave / (NSTRIPS_ * (NX_ / 16));

  // --- A operand: 16x4 f32, lanes 0-15 carry K=0,1; lanes 16-31 carry K=2,3.
  // Branchless: per-lane offset half*2 selects coord 0 or 2; coord 1 is
  // loaded unconditionally and zero-selected (K=3 pad) via cndmask.
  int m = mt * 16 + l;
  const float* xb = x + ((size_t)b * NX_ + m) * 3;
  float xa = xb[half * 2];
  float xm = xb[1];
  v2f A;
  A.x = -2.0f * xa;
  A.y = half ? 0.0f : (-2.0f * xm);

  // Per-row x-side terms for this lane's 8 C/D rows (M = r + 8*half):
  // 8 interleaved (sq, rs) pairs = 64 B contiguous per lane.
  float sqx_r[8], rsx_r[8];
  const float* wxb = wsx + 2 * ((size_t)b * NX_ + mt * 16 + half * 8);
#pragma unroll
  for (int r = 0; r < 8; ++r) {
    sqx_r[r] = wxb[2 * r];
    rsx_r[r] = wxb[2 * r + 1];
  }

  int n0 = ns * (16 * NT_) + l;
  for (int nt = 0; nt < NT_; ++nt) {
    int n = n0 + nt * 16;

    // --- B operand: 4x16 f32, mirrored layout (VGPR0: K=0|2, VGPR1: K=1|3).
    const float* yb = y + ((size_t)b * NY_ + n) * 3;
    float ya = yb[half * 2];
    float ym = yb[1];
    v2f Bv;
    Bv.x = ya;
    Bv.y = half ? 0.0f : ym;

    // y-side pair in one b64 load.
    const float* wyb = wsy + 2 * ((size_t)b * NY_ + n);
    float sqy_n = wyb[0];
    float rsy_n = wyb[1];

    v8f D = {};
    // D = A x B (+0): 16x16 tile of (-2 x.y), f32 accumulate.
    D = __builtin_amdgcn_wmma_f32_16x16x4_f32(
        /*neg_a=*/false, A, /*neg_b=*/false, Bv,
        /*c_mod=*/(short)0, D, /*reuse_a=*/false, /*reuse_b=*/false);

    float* ob = out + ((size_t)b * NX_ + (size_t)mt * 16 + half * 8) * NY_ + n;
#pragma unroll
    for (int r = 0; r < 8; ++r) {
      float d = D[r] + sqx_r[r] + sqy_n;                        // dist^2
      d = fminf(fmaxf(d, 1e-10f), 1e6f);
      float sxy = fminf(fmaxf(rsx_r[r] * rsy_n, 1e-5f), 1e6f);  // sqrt(sx*sy)
      float o = sxy * __builtin_amdgcn_rcpf(sxy + d);           // 1/(1+d/sxy)
      __builtin_nontemporal_store(o, ob + (size_t)r * NY_);     // streaming
    }
  }
}

extern "C" void kernel_launch(void* const* d_in, const int* in_sizes, int n_in,
                              void* d_out, int out_size, void* d_ws, size_t ws_size,
                              hipStream_t stream) {
  const float* x  = (const float*)d_in[0];   // (B,NX,3)
  const float* y  = (const float*)d_in[1];   // (B,NY,3)
  const float* sx = (const float*)d_in[2];   // (B,NX,F)
  const float* sy = (const float*)d_in[3];   // (B,NY,F)
  const float* sc = (const float*)d_in[4];   // (F,)
  float* out = (float*)d_out;                // (B,NX,NY)
  float* ws  = (float*)d_ws;                 // needs 4*BN_*4 = 256 KB

  (void)in_sizes; (void)n_in; (void)out_size; (void)ws_size;

  cauchy_precompute<<<BN_ / 256, 256, 0, stream>>>(x, y, sx, sy, sc, ws);

  int waves  = B_ * (NX_ / 16) * NSTRIPS_;   // 32768 waves
  int blocks = waves / 8;                    // 8 waves (256 thr) per block
  cauchy_wmma<<<blocks, 256, 0, stream>>>(x, y, ws, out);
}